// MPO_SquareLinearModel_73005854097733
// MI455X (gfx1250) — compile-verified
//
#include <hip/hip_runtime.h>
#include <hip/hip_bf16.h>

typedef float v2f __attribute__((ext_vector_type(2)));
typedef float v8f __attribute__((ext_vector_type(8)));

// ---------------------------------------------------------------------------
// WMMA wrapper: D = A(16x4 f32) x B(4x16 f32) + C(16x16 f32)
// ---------------------------------------------------------------------------
static __device__ __forceinline__ v8f wmma_16x16x4_f32(v2f a, v2f b, v8f c) {
#if __has_builtin(__builtin_amdgcn_wmma_f32_16x16x4_f32)
    return __builtin_amdgcn_wmma_f32_16x16x4_f32(
        /*neg_a=*/false, a, /*neg_b=*/false, b,
        /*c_mod=*/(short)0, c, /*reuse_a=*/false, /*reuse_b=*/false);
#else
    v8f d;
    asm volatile("v_wmma_f32_16x16x4_f32 %0, %1, %2, %3"
                 : "=v"(d)
                 : "v"(a), "v"(b), "v"(c));
    return d;
#endif
}

// ---------------------------------------------------------------------------
// Setup: contract the 8 MPO cores into W (256x256), stored directly in
// WMMA B-fragment order in d_ws.
//
// ws layout (floats):
//   [0       .. 262143]  ping buffer (T_k, up to 128*128*16)
//   [262144  .. 524287]  pong buffer
//   [524288  .. 589823]  W fragments (65536 floats = 256 KB)
//
// Fragment order (k-major, ntile-minor so DS offsets are immediates):
//   frag float2 index = (kstep*16 + ntile)*32 + lane, component v holds
//   W[K][N] with K = 4*kstep + (lane<16 ? 0 : 2) + v,
//              N = 16*ntile + (lane & 15)
// (B 4x16 layout: VGPR v, lanes 0-15 -> K=v, lanes 16-31 -> K=v+2.)
// ---------------------------------------------------------------------------
__global__ void mpo_setup_kernel(const float* __restrict__ w, float* __restrict__ ws) {
    float* bufA = ws;
    float* bufB = ws + 262144;
    float* wf   = ws + 524288;
    const int tid = threadIdx.x;          // 1024 threads

    // Step 1: T1[I,J,a] = core0[0,a,I,J] ; core0 flat idx = a*4 + I*2 + J
    if (tid < 64) {
        int a = tid & 15;
        int I = (tid >> 5) & 1;
        int J = (tid >> 4) & 1;
        bufA[(I * 2 + J) * 16 + a] = w[a * 4 + I * 2 + J];
    }
    __syncthreads();

    // Steps 2..7: T_{k+1}[(2I+i),(2J+j),b] = sum_a T_k[I,J,a] * core[a,b,i,j]
    int dim = 2;
    for (int c = 1; c <= 6; ++c) {
        const float* core = w + 64 + (c - 1) * 1024;   // shape (16,16,2,2)
        const int odim  = dim * 2;
        const int total = odim * odim * 16;
        for (int idx = tid; idx < total; idx += 1024) {
            int b   = idx & 15;
            int rem = idx >> 4;
            int Jo  = rem % odim;
            int Io  = rem / odim;
            int I = Io >> 1, i = Io & 1;
            int J = Jo >> 1, j = Jo & 1;
            const float* Trow = bufA + (I * dim + J) * 16;
            float s = 0.f;
#pragma unroll
            for (int a = 0; a < 16; ++a)
                s += Trow[a] * core[((a * 16 + b) * 2 + i) * 2 + j];
            bufB[idx] = s;
        }
        __syncthreads();
        float* t = bufA; bufA = bufB; bufB = t;        // 6 swaps -> T7 ends in ws[0..]
        dim = odim;
    }

    // Final: W[r][n] = sum_a T7[r>>1, n>>1, a] * core7[a, r&1, n&1]
    const float* core7 = w + 6208;                     // shape (16,1,2,2)
    for (int idx = tid; idx < 65536; idx += 1024) {
        int n = idx & 255;
        int r = idx >> 8;
        int I = r >> 1, i = r & 1;
        int J = n >> 1, j = n & 1;
        const float* Trow = bufA + (I * 128 + J) * 16;
        float s = 0.f;
#pragma unroll
        for (int a = 0; a < 16; ++a)
            s += Trow[a] * core7[a * 4 + i * 2 + j];
        // scatter into B-fragment layout (k-major, ntile-minor)
        int kstep = r >> 2, kk = r & 3;
        int nt    = n >> 4, nl = n & 15;
        int lane  = ((kk >= 2) ? 16 : 0) + nl;
        int v     = kk & 1;
        wf[((kstep * 16 + nt) * 32 + lane) * 2 + v] = s;
    }
}

// ---------------------------------------------------------------------------
// Main fused kernel.
// Block = 256 threads = 8 waves = 2 quads. Quad q owns a 64-row slab
// (4 strips of 16 rows); wave g within the quad owns n-tiles 4g..4g+3.
// Per wave: acc[strip][tile] = 4x4 WMMA tiles (128 VGPRs).
// Per k-step: 2 ds_load_2addr_b64 (4 B frags) + 4 global_load_b64 (A frags)
// feed 16 WMMAs. A fragments are software-pipelined one k-step ahead so the
// VMEM round trip is hidden under the previous step's WMMA stream (last step
// peeled to avoid prefetching past the end of x).
// Partials are combined across the quad through a small LDS buffer in a
// fixed order (deterministic, no float atomics).
// ---------------------------------------------------------------------------
__global__ void __launch_bounds__(256, 1)
mpo_main_kernel(const float* __restrict__ x,
                const float* __restrict__ dk,
                const float* __restrict__ bias,
                const float* __restrict__ wf,
                float* __restrict__ out) {
    extern __shared__ float lds[];     // 65536 W-frag floats + 512 partials
    float* part = lds + 65536;         // [q][g][s][16] = 2*4*4*16 floats
    const int tid = threadIdx.x;       // 256

    // cooperative W-fragment fill: 16384 float4 over 256 threads
    {
        const float4* src = (const float4*)wf;
        float4* dst = (float4*)lds;
        for (int i = tid; i < 16384; i += 256)
            dst[i] = src[i];
    }
    __syncthreads();

    const int wave = tid >> 5;
    const int lane = tid & 31;
    const int l16  = lane & 15;
    const int koff = (lane < 16) ? 0 : 2;
    const int q    = wave >> 2;        // row-slab quad (0..1)
    const int g    = wave & 3;         // n-tile group (0..3)

    const long rowbase = ((long)blockIdx.x * 2 + q) * 64;   // slab's first row
    const float* xbase = x + (rowbase + l16) * 256 + koff;

    v8f acc[4][4];
#pragma unroll
    for (int s = 0; s < 4; ++s)
#pragma unroll
        for (int t = 0; t < 4; ++t)
#pragma unroll
            for (int v = 0; v < 8; ++v)
                acc[s][t][v] = 0.f;

    // B frags for this wave's 4 tiles sit 256 B apart at g*1024 B; k stride
    // 4096 B. A frags: strip stride = 16 rows * 1 KB = 16384 B immediates.
    const float* ldsb = lds + lane * 2 + g * 256;

    // prologue: load A fragments for k = 0
    v2f a0 = *(const v2f*)(xbase);
    v2f a1 = *(const v2f*)(xbase + 4096);
    v2f a2 = *(const v2f*)(xbase + 8192);
    v2f a3 = *(const v2f*)(xbase + 12288);

    for (int k = 0; k < 63; ++k) {
        // prefetch A for k+1 (one full WMMA block of lead time)
        const float* xn = xbase + 4 * (k + 1);
        v2f n0 = *(const v2f*)(xn);
        v2f n1 = *(const v2f*)(xn + 4096);
        v2f n2 = *(const v2f*)(xn + 8192);
        v2f n3 = *(const v2f*)(xn + 12288);

        const float* lk = ldsb + k * 1024;
        v2f b0 = *(const v2f*)(lk);
        v2f b1 = *(const v2f*)(lk + 64);
        v2f b2 = *(const v2f*)(lk + 128);
        v2f b3 = *(const v2f*)(lk + 192);

        acc[0][0] = wmma_16x16x4_f32(a0, b0, acc[0][0]);
        acc[0][1] = wmma_16x16x4_f32(a0, b1, acc[0][1]);
        acc[0][2] = wmma_16x16x4_f32(a0, b2, acc[0][2]);
        acc[0][3] = wmma_16x16x4_f32(a0, b3, acc[0][3]);
        acc[1][0] = wmma_16x16x4_f32(a1, b0, acc[1][0]);
        acc[1][1] = wmma_16x16x4_f32(a1, b1, acc[1][1]);
        acc[1][2] = wmma_16x16x4_f32(a1, b2, acc[1][2]);
        acc[1][3] = wmma_16x16x4_f32(a1, b3, acc[1][3]);
        acc[2][0] = wmma_16x16x4_f32(a2, b0, acc[2][0]);
        acc[2][1] = wmma_16x16x4_f32(a2, b1, acc[2][1]);
        acc[2][2] = wmma_16x16x4_f32(a2, b2, acc[2][2]);
        acc[2][3] = wmma_16x16x4_f32(a2, b3, acc[2][3]);
        acc[3][0] = wmma_16x16x4_f32(a3, b0, acc[3][0]);
        acc[3][1] = wmma_16x16x4_f32(a3, b1, acc[3][1]);
        acc[3][2] = wmma_16x16x4_f32(a3, b2, acc[3][2]);
        acc[3][3] = wmma_16x16x4_f32(a3, b3, acc[3][3]);

        a0 = n0; a1 = n1; a2 = n2; a3 = n3;
    }
    // peeled final step k = 63 (no prefetch -> no OOB read past x)
    {
        const float* lk = ldsb + 63 * 1024;
        v2f b0 = *(const v2f*)(lk);
        v2f b1 = *(const v2f*)(lk + 64);
        v2f b2 = *(const v2f*)(lk + 128);
        v2f b3 = *(const v2f*)(lk + 192);
        acc[0][0] = wmma_16x16x4_f32(a0, b0, acc[0][0]);
        acc[0][1] = wmma_16x16x4_f32(a0, b1, acc[0][1]);
        acc[0][2] = wmma_16x16x4_f32(a0, b2, acc[0][2]);
        acc[0][3] = wmma_16x16x4_f32(a0, b3, acc[0][3]);
        acc[1][0] = wmma_16x16x4_f32(a1, b0, acc[1][0]);
        acc[1][1] = wmma_16x16x4_f32(a1, b1, acc[1][1]);
        acc[1][2] = wmma_16x16x4_f32(a1, b2, acc[1][2]);
        acc[1][3] = wmma_16x16x4_f32(a1, b3, acc[1][3]);
        acc[2][0] = wmma_16x16x4_f32(a2, b0, acc[2][0]);
        acc[2][1] = wmma_16x16x4_f32(a2, b1, acc[2][1]);
        acc[2][2] = wmma_16x16x4_f32(a2, b2, acc[2][2]);
        acc[2][3] = wmma_16x16x4_f32(a2, b3, acc[2][3]);
        acc[3][0] = wmma_16x16x4_f32(a3, b0, acc[3][0]);
        acc[3][1] = wmma_16x16x4_f32(a3, b1, acc[3][1]);
        acc[3][2] = wmma_16x16x4_f32(a3, b2, acc[3][2]);
        acc[3][3] = wmma_16x16x4_f32(a3, b3, acc[3][3]);
    }

    // Epilogue: square, scale by dense_kernel, reduce over this wave's tiles
    float kv[4];
#pragma unroll
    for (int t = 0; t < 4; ++t)
        kv[t] = dk[(g * 4 + t) * 16 + l16];

#pragma unroll
    for (int s = 0; s < 4; ++s) {
        float sv[8];
#pragma unroll
        for (int v = 0; v < 8; ++v) sv[v] = 0.f;
#pragma unroll
        for (int t = 0; t < 4; ++t)
#pragma unroll
            for (int v = 0; v < 8; ++v) {
                float h = acc[s][t][v];
                sv[v] += h * h * kv[t];
            }
        // reduce across the 16-lane N group
#pragma unroll
        for (int v = 0; v < 8; ++v) {
            sv[v] += __shfl_xor(sv[v], 1, 32);
            sv[v] += __shfl_xor(sv[v], 2, 32);
            sv[v] += __shfl_xor(sv[v], 4, 32);
            sv[v] += __shfl_xor(sv[v], 8, 32);
        }
        if (l16 == 0) {
            int half = (lane >= 16) ? 8 : 0;   // lanes 0-15: rows v; 16-31: v+8
            float* p = part + ((q * 4 + g) * 4 + s) * 16 + half;
#pragma unroll
            for (int v = 0; v < 8; ++v) p[v] = sv[v];
        }
    }
    __syncthreads();

    // Deterministic combine: row_local = tid = q2*64 + s2*16 + idx
    if (tid < 128) {
        int q2  = tid >> 6;
        int s2  = (tid >> 4) & 3;
        int idx = tid & 15;
        const float* p = part + (q2 * 16 + s2) * 16 + idx;  // g stride = 64
        float r = p[0] + p[64] + p[128] + p[192];
        out[(long)blockIdx.x * 128 + tid] = r + bias[0];
    }
}

// ---------------------------------------------------------------------------
extern "C" void kernel_launch(void* const* d_in, const int* in_sizes, int n_in,
                              void* d_out, int out_size, void* d_ws, size_t ws_size,
                              hipStream_t stream) {
    const float* x   = (const float*)d_in[0];   // (262144, 256)
    const float* mpw = (const float*)d_in[1];   // (6272,)
    const float* dk  = (const float*)d_in[2];   // (256, 1)
    const float* db  = (const float*)d_in[3];   // (1,)
    float* out = (float*)d_out;                 // (262144, 1)
    float* ws  = (float*)d_ws;

    // 1) contract MPO cores -> W in B-fragment layout (ws + 524288 floats)
    hipLaunchKernelGGL(mpo_setup_kernel, dim3(1), dim3(1024), 0, stream, mpw, ws);

    // 2) fused GEMM + square + weighted row-reduction
    //    262144 rows / (128 rows per block) = 2048 blocks
    //    dynamic LDS = 256 KB W fragments + 2 KB partials
    hipLaunchKernelGGL(mpo_main_kernel, dim3(2048), dim3(256),
                       (65536 + 512) * sizeof(float), stream,
                       x, dk, db, ws + 524288, out);
}